// FFTConvNet_39444979647007
// MI455X (gfx1250) — compile-verified
//
#include <hip/hip_runtime.h>
#include <math.h>

// ---------------------------------------------------------------------------
// FFTConvNet on gfx1250:  FFT2D -> per-frequency complex GEMM (WMMA f32) ->
// IFFT2D + bias + ReLU.   B=32, CIN=COUT=64, H=W=128.
// ---------------------------------------------------------------------------

#define BATCH 32
#define CIN   64
#define COUT  64
#define NPIX  16384      // 128*128
#define PIT   129        // LDS row pitch (odd -> conflict-free column pass)
#define NF    8          // frequencies per workgroup in the GEMM kernel

typedef float v2f __attribute__((ext_vector_type(2)));
typedef float v8f __attribute__((ext_vector_type(8)));

__device__ __forceinline__ unsigned brev7(unsigned x) { return __brev(x) >> 25u; }

// In-LDS 128x128 radix-2 2D FFT.  Input must be loaded with the W (fast)
// dimension bit-reversed.  sgn = -1 forward, +1 inverse.
__device__ __forceinline__ void fft2d_inlds(float* sre, float* sim, int tid, float sgn) {
    // ---- row FFTs (128 rows, 64 butterflies each per stage) ----
    for (int s = 1; s <= 7; ++s) {
        const int m = 1 << s, half = m >> 1;
        const float w0 = sgn * 6.28318530717958647f / (float)m;
        __syncthreads();
        for (int idx = tid; idx < 8192; idx += 256) {
            int line = idx >> 6, bi = idx & 63;
            int j = bi & (half - 1), g = bi >> (s - 1);
            int a0 = line * PIT + g * m + j, a1 = a0 + half;
            float sw, cw; __sincosf(w0 * (float)j, &sw, &cw);
            float pr = sre[a1], pi = sim[a1];
            float tr = cw * pr - sw * pi, ti = cw * pi + sw * pr;
            float ur = sre[a0], ui = sim[a0];
            sre[a0] = ur + tr; sim[a0] = ui + ti;
            sre[a1] = ur - tr; sim[a1] = ui - ti;
        }
    }
    __syncthreads();
    // ---- bit-reverse rows (prepares DIT along H) ----
    for (int i = tid; i < NPIX; i += 256) {
        int h = i >> 7, w = i & 127, rh = (int)brev7((unsigned)h);
        if (h < rh) {
            int a = h * PIT + w, b = rh * PIT + w;
            float t = sre[a]; sre[a] = sre[b]; sre[b] = t;
            t = sim[a]; sim[a] = sim[b]; sim[b] = t;
        }
    }
    // ---- column FFTs ----
    for (int s = 1; s <= 7; ++s) {
        const int m = 1 << s, half = m >> 1;
        const float w0 = sgn * 6.28318530717958647f / (float)m;
        __syncthreads();
        for (int idx = tid; idx < 8192; idx += 256) {
            int col = idx >> 6, bi = idx & 63;
            int j = bi & (half - 1), g = bi >> (s - 1);
            int a0 = (g * m + j) * PIT + col, a1 = a0 + half * PIT;
            float sw, cw; __sincosf(w0 * (float)j, &sw, &cw);
            float pr = sre[a1], pi = sim[a1];
            float tr = cw * pr - sw * pi, ti = cw * pi + sw * pr;
            float ur = sre[a0], ui = sim[a0];
            sre[a0] = ur + tr; sim[a0] = ui + ti;
            sre[a1] = ur - tr; sim[a1] = ui - ti;
        }
    }
    __syncthreads();
}

// K1: forward FFT of one (b,c) image per workgroup.  x real -> xf complex.
__global__ void __launch_bounds__(256) fwd_fft_kernel(const float* __restrict__ x,
                                                      float2* __restrict__ xf) {
    __shared__ float sre[128 * PIT];
    __shared__ float sim[128 * PIT];
    const int tid = threadIdx.x;
    const size_t img = blockIdx.x;
    const float* src = x + img * NPIX;
    for (int i = tid; i < NPIX; i += 256) {
        int h = i >> 7, w = i & 127;
        int a = h * PIT + (int)brev7((unsigned)w);
        sre[a] = src[i];
        sim[a] = 0.0f;
    }
    fft2d_inlds(sre, sim, tid, -1.0f);
    float2* dst = xf + img * NPIX;
    for (int i = tid; i < NPIX; i += 256) {
        int h = i >> 7, w = i & 127;
        dst[i] = make_float2(sre[h * PIT + w], sim[h * PIT + w]);
    }
}

// K2: per-frequency complex GEMM, Y[b,o,f] = sum_c X[b,c,f] * W[o,c,f].
// Workgroup: 8 waves = 8 consecutive frequencies; each wave owns one
// frequency and its full 32x64 complex output via v_wmma_f32_16x16x4_f32.
// Output goes through LDS so global stores are 64B-contiguous per (b,o).
__global__ void __launch_bounds__(256) freq_gemm_wmma(const float2* __restrict__ xf,
                                                      const float* __restrict__ wre,
                                                      const float* __restrict__ wim,
                                                      float2* __restrict__ yf) {
    // 144 KB LDS, used in two phases:
    //  phase A (K streaming): X/W staging arrays, f dim padded to 9.
    //  phase B (output):      float2 yst[(b*64+o)*9 + f]  (pad 9 -> no conflicts)
    __shared__ __align__(16) float smem[36864];
    float (*lxr)[BATCH][9] = (float (*)[BATCH][9])(smem + 0);      // X real  [c][b][f]
    float (*lxi)[BATCH][9] = (float (*)[BATCH][9])(smem + 4608);   // X imag
    float (*lxn)[BATCH][9] = (float (*)[BATCH][9])(smem + 9216);   // -X imag (f32 WMMA has no A-neg)
    float (*lwr)[COUT][9]  = (float (*)[COUT][9])(smem + 13824);   // W real  [c][o][f]
    float (*lwi)[COUT][9]  = (float (*)[COUT][9])(smem + 23040);   // W imag

    const int tid  = threadIdx.x;
    const int lane = tid & 31;
    const int wid  = tid >> 5;            // wave id == local frequency
    const int fb   = blockIdx.x * NF;     // base frequency
    const int lb   = lane & 15;           // M / N index within tile
    const int chi  = (lane >> 4) << 1;    // K offset 0 (lanes 0-15) / 2 (16-31)

    v8f accR[2][4], accI[2][4];
#pragma unroll
    for (int mt = 0; mt < 2; ++mt)
#pragma unroll
        for (int nt = 0; nt < 4; ++nt) { accR[mt][nt] = (v8f)0.0f; accI[mt][nt] = (v8f)0.0f; }

    for (int cc = 0; cc < 4; ++cc) {
        __syncthreads();
        // stage X chunk: 16c x 32b x 8f  (f-contiguous global reads)
        for (int t = tid; t < 16 * BATCH * NF; t += 256) {
            int f = t & 7, r = t >> 3;
            int b = r & 31, c = r >> 5;
            float2 v = xf[(size_t)(b * CIN + cc * 16 + c) * NPIX + fb + f];
            lxr[c][b][f] = v.x;
            lxi[c][b][f] = v.y;
            lxn[c][b][f] = -v.y;
        }
        // stage W chunk: 16c x 64o x 8f
        for (int t = tid; t < 16 * COUT * NF; t += 256) {
            int f = t & 7, r = t >> 3;
            int o = r & 63, c = r >> 6;
            size_t g = (size_t)(o * CIN + cc * 16 + c) * NPIX + fb + f;
            lwr[c][o][f] = wre[g];
            lwi[c][o][f] = wim[g];
        }
        __syncthreads();

#pragma unroll
        for (int k4 = 0; k4 < 4; ++k4) {
            const int c0 = k4 * 4 + chi;  // this lane-half's K pair
            // A fragments (16x4, M=batch): v0=K even, v1=K odd per lane half
            v2f ar[2], ai[2], an[2];
#pragma unroll
            for (int mt = 0; mt < 2; ++mt) {
                int b = mt * 16 + lb;
                ar[mt] = (v2f){lxr[c0][b][wid], lxr[c0 + 1][b][wid]};
                ai[mt] = (v2f){lxi[c0][b][wid], lxi[c0 + 1][b][wid]};
                an[mt] = (v2f){lxn[c0][b][wid], lxn[c0 + 1][b][wid]};
            }
            // B fragments (4x16, N=cout): B[k][n] = W[o=n][c0+k] (transposed)
            v2f br_[4], bi_[4];
#pragma unroll
            for (int nt = 0; nt < 4; ++nt) {
                int o = nt * 16 + lb;
                br_[nt] = (v2f){lwr[c0][o][wid], lwr[c0 + 1][o][wid]};
                bi_[nt] = (v2f){lwi[c0][o][wid], lwi[c0 + 1][o][wid]};
            }
#pragma unroll
            for (int mt = 0; mt < 2; ++mt)
#pragma unroll
                for (int nt = 0; nt < 4; ++nt) {
                    // real += Xr*Wr + (-Xi)*Wi ; imag += Xr*Wi + Xi*Wr
                    accR[mt][nt] = __builtin_amdgcn_wmma_f32_16x16x4_f32(
                        false, ar[mt], false, br_[nt], (short)0, accR[mt][nt], false, false);
                    accR[mt][nt] = __builtin_amdgcn_wmma_f32_16x16x4_f32(
                        false, an[mt], false, bi_[nt], (short)0, accR[mt][nt], false, false);
                    accI[mt][nt] = __builtin_amdgcn_wmma_f32_16x16x4_f32(
                        false, ar[mt], false, bi_[nt], (short)0, accI[mt][nt], false, false);
                    accI[mt][nt] = __builtin_amdgcn_wmma_f32_16x16x4_f32(
                        false, ai[mt], false, br_[nt], (short)0, accI[mt][nt], false, false);
                }
        }
    }

    // ---- phase B: stage results in LDS, then 64B-contiguous global stores ----
    __syncthreads();                       // staging arrays are dead now
    float2* yst = (float2*)smem;           // yst[(b*64+o)*9 + f], 18432 float2
    // D layout: VGPR v, lanes 0-15 -> M=v ; lanes 16-31 -> M=v+8 ; N=lane%16
    const int mhi = (lane >> 4) << 3;
#pragma unroll
    for (int mt = 0; mt < 2; ++mt)
#pragma unroll
        for (int nt = 0; nt < 4; ++nt)
#pragma unroll
            for (int v = 0; v < 8; ++v) {
                int b = mt * 16 + v + mhi;
                int o = nt * 16 + lb;
                yst[(b * COUT + o) * 9 + wid] =
                    make_float2(accR[mt][nt][v], accI[mt][nt][v]);
            }
    __syncthreads();
    // each (b,o): 8 consecutive float2 (64B, 64B-aligned since fb%8==0)
    for (int t = tid; t < NF * BATCH * COUT; t += 256) {
        int f = t & 7, bo = t >> 3;
        yf[(size_t)bo * NPIX + fb + f] = yst[bo * 9 + f];
    }
}

// K3: inverse FFT of one (b,o) image + 1/N^2 scale + bias + ReLU, real part.
__global__ void __launch_bounds__(256) inv_fft_bias_relu(const float2* __restrict__ yf,
                                                         const float* __restrict__ bias,
                                                         float* __restrict__ out) {
    __shared__ float sre[128 * PIT];
    __shared__ float sim[128 * PIT];
    const int tid = threadIdx.x;
    const size_t img = blockIdx.x;
    const int och = (int)(img & 63);
    const float2* src = yf + img * NPIX;
    for (int i = tid; i < NPIX; i += 256) {
        int h = i >> 7, w = i & 127;
        int a = h * PIT + (int)brev7((unsigned)w);
        float2 v = src[i];
        sre[a] = v.x;
        sim[a] = v.y;
    }
    fft2d_inlds(sre, sim, tid, +1.0f);
    const float bo = bias[och];
    float* dst = out + img * NPIX;
    for (int i = tid; i < NPIX; i += 256) {
        int h = i >> 7, w = i & 127;
        float v = fmaf(sre[h * PIT + w], 1.0f / 16384.0f, bo);
        dst[i] = fmaxf(v, 0.0f);
    }
}

extern "C" void kernel_launch(void* const* d_in, const int* in_sizes, int n_in,
                              void* d_out, int out_size, void* d_ws, size_t ws_size,
                              hipStream_t stream) {
    const float* x    = (const float*)d_in[0];  // (32,64,128,128)
    const float* wre  = (const float*)d_in[1];  // (64,64,128,128)
    const float* wim  = (const float*)d_in[2];  // (64,64,128,128)
    const float* bias = (const float*)d_in[3];  // (64,)
    float* out = (float*)d_out;

    // workspace: XF (268 MB) | YF (268 MB)
    float2* xfbuf = (float2*)d_ws;
    float2* yfbuf = (float2*)((char*)d_ws + (size_t)BATCH * CIN * NPIX * sizeof(float2));

    fwd_fft_kernel<<<BATCH * CIN, 256, 0, stream>>>(x, xfbuf);
    freq_gemm_wmma<<<NPIX / NF, 256, 0, stream>>>(xfbuf, wre, wim, yfbuf);
    inv_fft_bias_relu<<<BATCH * COUT, 256, 0, stream>>>(yfbuf, bias, out);
}